// SelfAttentiveSpanPooler_26817775796243
// MI455X (gfx1250) — compile-verified
//
#include <hip/hip_runtime.h>
#include <hip/hip_bf16.h>
#include <stdint.h>

// Problem constants (match reference)
#define B_   8
#define S_   4096
#define D_   1024
#define NSP  512
#define SPW  16

typedef __attribute__((ext_vector_type(2))) float v2f;
typedef __attribute__((ext_vector_type(8))) float v8f;

// One workgroup per span. 256 threads = 8 waves (wave32).
__global__ __launch_bounds__(256) void span_pool_kernel(
    const float* __restrict__ h,          // [B, S, D]
    const long long* __restrict__ spans,  // [B, NSP, SPW] int64, -1 = pad
    const float* __restrict__ w,          // [D]
    float* __restrict__ out)              // [B, NSP, D]
{
    __shared__ float Hs[SPW * D_];   // 64 KB: gathered span rows
    __shared__ int   rowsrc[SPW];
    __shared__ int   maskw[SPW];
    __shared__ float att_s[SPW];
    __shared__ float wts_s[SPW];

    const int tid  = threadIdx.x;
    const int lane = tid & 31;
    const int wv   = tid >> 5;            // wave id 0..7
    const int sp   = blockIdx.x;          // span id in [0, B*NSP)
    const int b    = sp >> 9;             // sp / NSP

    // ---- phase 0: read span indices ----
    if (tid < SPW) {
        long long v = spans[(size_t)sp * SPW + tid];
        int m = (v == -1);
        maskw[tid]  = m;
        rowsrc[tid] = m ? 0 : (int)v;     // masked rows read row 0 (weight 0)
    }
    __syncthreads();

    // ---- phase 1: async gather of 16 rows into LDS (CDNA5 async path) ----
    // Iteration i loads row i completely: 256 lanes x 16B = 4 KB = one row.
    {
        uint32_t lds_base = (uint32_t)(size_t)(void*)&Hs[0];  // low 32 bits = LDS offset
        #pragma unroll
        for (int i = 0; i < SPW; ++i) {
            int row = rowsrc[i];
            const float* src = h + ((size_t)b * S_ + (size_t)row) * D_ + tid * 4;
            uint32_t loff = lds_base + (uint32_t)(i * D_ + tid * 4) * 4u;
            asm volatile("global_load_async_to_lds_b128 %0, %1, off"
                         :: "v"(loff), "v"(src) : "memory");
        }
        asm volatile("s_wait_asynccnt 0x0" ::: "memory");
    }
    __syncthreads();

    // ---- phase 2: logits = Hs . w  (2 rows per wave, shuffle reduction) ----
    #pragma unroll
    for (int rr = 0; rr < 2; ++rr) {
        int row = wv * 2 + rr;
        float acc = 0.0f;
        #pragma unroll 8
        for (int c = lane; c < D_; c += 32)
            acc += Hs[row * D_ + c] * w[c];
        #pragma unroll
        for (int m = 16; m >= 1; m >>= 1)
            acc += __shfl_xor(acc, m, 32);
        if (lane == 0)
            att_s[row] = maskw[row] ? -10000.0f : acc;
    }
    __syncthreads();

    // ---- phase 3: masked softmax over span width (16 lanes, trivial) ----
    if (tid < SPW) {
        float mx = -3.4e38f;
        #pragma unroll
        for (int j = 0; j < SPW; ++j) mx = fmaxf(mx, att_s[j]);
        float ss = 0.0f;
        #pragma unroll
        for (int j = 0; j < SPW; ++j) ss += expf(att_s[j] - mx);
        wts_s[tid] = maskw[tid] ? 0.0f : (expf(att_s[tid] - mx) / ss);
    }
    __syncthreads();

    // ---- phase 4: weighted sum via V_WMMA_F32_16X16X4_F32 ----
    // Per 16-dim tile: C(16x16) += A(16x4) x B(4x16), 4 K-chunks cover width 16.
    //   A[m,k] = Hs[kbase+k][d0+m]   (rows = dims, cols = width slots)
    //   B[k,n] = wts[kbase+k]        (broadcast along N)
    // => C[m,n] = out[d0+m], replicated over N.
    const int hi  = lane >> 4;   // 0: lanes 0-15, 1: lanes 16-31
    const int lm  = lane & 15;
    const int hi2 = hi * 2;

    // B operands are tile-invariant: hoist. VGPR0={K0|K2}, VGPR1={K1|K3}.
    v2f bk[4];
    #pragma unroll
    for (int j = 0; j < 4; ++j) {
        bk[j].x = wts_s[4 * j + hi2];
        bk[j].y = wts_s[4 * j + 1 + hi2];
    }

    float* outp = out + (size_t)sp * D_;
    #pragma unroll
    for (int t = 0; t < 8; ++t) {
        int d0 = (wv * 8 + t) * 16;       // 8 waves x 8 tiles = 64 tiles = D
        v8f c = {};
        #pragma unroll
        for (int j = 0; j < 4; ++j) {
            v2f a;
            a.x = Hs[(4 * j + hi2) * D_ + d0 + lm];       // A VGPR0: K = kbase + 2*hi
            a.y = Hs[(4 * j + 1 + hi2) * D_ + d0 + lm];   // A VGPR1: K = kbase + 1 + 2*hi
            c = __builtin_amdgcn_wmma_f32_16x16x4_f32(
                    false, a, false, bk[j], (short)0, c, false, false);
        }
        // Lane 0 holds out[d0..d0+7] in c[0..7]; lane 16 holds out[d0+8..d0+15].
        if (lm == 0) {
            int dbase = d0 + hi * 8;
            float4 lo4 = make_float4(c[0], c[1], c[2], c[3]);
            float4 hi4 = make_float4(c[4], c[5], c[6], c[7]);
            *(float4*)(outp + dbase)     = lo4;
            *(float4*)(outp + dbase + 4) = hi4;
        }
    }
}

extern "C" void kernel_launch(void* const* d_in, const int* in_sizes, int n_in,
                              void* d_out, int out_size, void* d_ws, size_t ws_size,
                              hipStream_t stream) {
    const float*     h     = (const float*)d_in[0];
    const long long* spans = (const long long*)d_in[1];
    const float*     w     = (const float*)d_in[2];
    float*           out   = (float*)d_out;

    dim3 grid(B_ * NSP);   // 4096 workgroups, one per span
    dim3 block(256);       // 8 wave32 waves
    hipLaunchKernelGGL(span_pool_kernel, grid, block, 0, stream, h, spans, w, out);
}